// SimpleGCN_16724602651053
// MI455X (gfx1250) — compile-verified
//
#include <hip/hip_runtime.h>
#include <hip/hip_bf16.h>

typedef float v2f __attribute__((ext_vector_type(2)));
typedef float v8f __attribute__((ext_vector_type(8)));

#define N_NODES 50000
#define N_EDGES 800000
#define DIM_IN  128
#define DIM_OUT 64

// ---------------- init: acc = 0, deg = 1 (self-loop contribution) --------------
__global__ __launch_bounds__(256) void gcn_init(float* __restrict__ acc,
                                                float* __restrict__ deg) {
    int i = blockIdx.x * blockDim.x + threadIdx.x;
    if (i < N_NODES * DIM_OUT) acc[i] = 0.0f;
    if (i < N_NODES)           deg[i] = 1.0f;
}

// ---------------- GEMM: h = x @ W via V_WMMA_F32_16X16X4_F32 -------------------
// Block = 128 threads = 4 waves. Wave w computes the 16x16 tile at
// (m0 = blockIdx.x*16, n0 = w*16). K = 128 -> 32 WMMA steps of K=4.
// A 16x4 layout (ISA 7.12.2): lane L -> row M=L&15, VGPR0/1 = K = 2*(L>>4)+{0,1}.
// B 4x16: lane L -> col N=L&15, VGPR0/1 = same K pattern.
// C/D 16x16: VGPR v -> row v + 8*(L>>4), col L&15.
__global__ __launch_bounds__(128) void gcn_gemm(const float* __restrict__ x,
                                                const float* __restrict__ W,
                                                float* __restrict__ h) {
    const int wave = threadIdx.x >> 5;
    const int lane = threadIdx.x & 31;
    const int m0   = blockIdx.x * 16;          // 50000/16 = 3125 exactly
    const int n0   = wave * 16;
    const int mrow = m0 + (lane & 15);
    const int kofs = (lane >> 4) * 2;
    const int ncol = n0 + (lane & 15);

    v8f c = {};
    #pragma unroll
    for (int k0 = 0; k0 < DIM_IN; k0 += 4) {
        v2f a = *(const v2f*)(x + (size_t)mrow * DIM_IN + (k0 + kofs));
        v2f b;
        b.x = W[(size_t)(k0 + kofs)     * DIM_OUT + ncol];
        b.y = W[(size_t)(k0 + kofs + 1) * DIM_OUT + ncol];
        c = __builtin_amdgcn_wmma_f32_16x16x4_f32(
                /*neg_a=*/false, a, /*neg_b=*/false, b,
                /*c_mod=*/(short)0, c, /*reuse_a=*/false, /*reuse_b=*/false);
    }

    const int rbase = m0 + ((lane >> 4) << 3);
    #pragma unroll
    for (int v = 0; v < 8; ++v)
        h[(size_t)(rbase + v) * DIM_OUT + ncol] = c[v];
}

// ---------------- degree: deg[dst] += 1 per edge -------------------------------
__global__ __launch_bounds__(256) void gcn_degree(const int* __restrict__ dst,
                                                  float* __restrict__ deg) {
    int e = blockIdx.x * blockDim.x + threadIdx.x;
    if (e < N_EDGES) atomicAdd(&deg[dst[e]], 1.0f);
}

// ---------------- dinv = deg > 0 ? rsqrt(deg) : 0 (in place) -------------------
__global__ __launch_bounds__(256) void gcn_dinv(float* __restrict__ deg) {
    int i = blockIdx.x * blockDim.x + threadIdx.x;
    if (i < N_NODES) {
        float d = deg[i];
        deg[i] = (d > 0.0f) ? rsqrtf(d) : 0.0f;
    }
}

// ---------------- edge aggregation: acc[dst] += h[src] * norm ------------------
// One wave per edge; each lane owns 2 columns (float2 gather + 2 f32 atomics).
// h (12.8MB) and acc (12.8MB) are L2-resident (192MB L2) -> atomics stay in L2.
__global__ __launch_bounds__(256) void gcn_aggregate(const int* __restrict__ src,
                                                     const int* __restrict__ dst,
                                                     const float* __restrict__ h,
                                                     const float* __restrict__ dinv,
                                                     float* __restrict__ acc) {
    long long tid = (long long)blockIdx.x * blockDim.x + threadIdx.x;
    int e    = (int)(tid >> 5);
    int lane = (int)(tid & 31);
    if (e >= N_EDGES) return;
    int s = src[e];
    int d = dst[e];
    float norm = dinv[s] * dinv[d];
    v2f hv = *(const v2f*)(h + (size_t)s * DIM_OUT + lane * 2);
    float* o = acc + (size_t)d * DIM_OUT + lane * 2;
    atomicAdd(o,     hv.x * norm);
    atomicAdd(o + 1, hv.y * norm);
}

// ---------------- finalize: +self-loop, +bias, row log_softmax -----------------
// One wave per node; lane owns columns (lane, lane+32); wave32 shuffle reduce.
__global__ __launch_bounds__(256) void gcn_finalize(const float* __restrict__ acc,
                                                    const float* __restrict__ h,
                                                    const float* __restrict__ dinv,
                                                    const float* __restrict__ bias,
                                                    float* __restrict__ out) {
    int tid  = blockIdx.x * blockDim.x + threadIdx.x;
    int n    = tid >> 5;
    int lane = tid & 31;
    if (n >= N_NODES) return;

    float sn = dinv[n];
    sn = sn * sn;                                  // self-loop norm = dinv^2
    size_t base = (size_t)n * DIM_OUT;
    float v0 = acc[base + lane]      + h[base + lane]      * sn + bias[lane];
    float v1 = acc[base + lane + 32] + h[base + lane + 32] * sn + bias[lane + 32];

    float m = fmaxf(v0, v1);
    #pragma unroll
    for (int off = 16; off > 0; off >>= 1)
        m = fmaxf(m, __shfl_xor(m, off, 32));

    float s = __expf(v0 - m) + __expf(v1 - m);
    #pragma unroll
    for (int off = 16; off > 0; off >>= 1)
        s += __shfl_xor(s, off, 32);

    float lse = m + __logf(s);
    out[base + lane]      = v0 - lse;
    out[base + lane + 32] = v1 - lse;
}

// -------------------------------------------------------------------------------
extern "C" void kernel_launch(void* const* d_in, const int* in_sizes, int n_in,
                              void* d_out, int out_size, void* d_ws, size_t ws_size,
                              hipStream_t stream) {
    const float* x    = (const float*)d_in[0];                 // [N, 128]
    const int*   ei   = (const int*)  d_in[1];                 // [2, E]
    const float* W    = (const float*)d_in[2];                 // [128, 64]
    const float* bias = (const float*)d_in[3];                 // [64]
    float* out = (float*)d_out;                                // [N, 64]

    const int* src = ei;                                       // edge_index[0]
    const int* dst = ei + N_EDGES;                             // edge_index[1]

    // workspace layout: h [N*64] | acc [N*64] | deg/dinv [N]
    float* h   = (float*)d_ws;
    float* acc = h   + (size_t)N_NODES * DIM_OUT;
    float* deg = acc + (size_t)N_NODES * DIM_OUT;

    {   // init acc + deg (covers max(N*64, N) elements)
        int total  = N_NODES * DIM_OUT;
        int blocks = (total + 255) / 256;
        gcn_init<<<blocks, 256, 0, stream>>>(acc, deg);
    }
    {   // WMMA GEMM: 3125 blocks x (4 waves = 4 column tiles)
        gcn_gemm<<<N_NODES / 16, 128, 0, stream>>>(x, W, h);
    }
    {   // degree atomics
        gcn_degree<<<(N_EDGES + 255) / 256, 256, 0, stream>>>(dst, deg);
    }
    {   // dinv
        gcn_dinv<<<(N_NODES + 255) / 256, 256, 0, stream>>>(deg);
    }
    {   // edge aggregation: one wave per edge
        long long threads = (long long)N_EDGES * 32;
        int blocks = (int)((threads + 255) / 256);
        gcn_aggregate<<<blocks, 256, 0, stream>>>(src, dst, h, deg, acc);
    }
    {   // finalize: one wave per node
        long long threads = (long long)N_NODES * 32;
        int blocks = (int)((threads + 255) / 256);
        gcn_finalize<<<blocks, 256, 0, stream>>>(acc, h, deg, bias, out);
    }
}